// DynamicGraphStorage_11897059410433
// MI455X (gfx1250) — compile-verified
//
#include <hip/hip_runtime.h>
#include <math.h>

#define HID 128
#define NEG_SLOPE 0.2f
#define GAT_EPS 1e-16f

typedef __attribute__((ext_vector_type(2))) float v2f;
typedef __attribute__((ext_vector_type(8))) float v8f;

// ---- monotone float <-> u32 encoding for atomic segment-max ----
__device__ __forceinline__ unsigned fkey(float x) {
  unsigned u = __float_as_uint(x);
  return (u & 0x80000000u) ? ~u : (u | 0x80000000u);
}
__device__ __forceinline__ float finv(unsigned u) {
  return __uint_as_float((u & 0x80000000u) ? (u ^ 0x80000000u) : ~u);
}

// K0: out = broadcast(bias); zero per-node accumulators
__global__ void k_init(float* __restrict__ out, const float* __restrict__ bias,
                       float* s_edge, float* deg, float* denom, unsigned* amax, int N) {
  long long idx = (long long)blockIdx.x * blockDim.x + threadIdx.x;
  long long total = (long long)N * HID;
  if (idx < total) out[idx] = bias[idx & (HID - 1)];
  if (idx < N) { s_edge[idx] = 0.f; deg[idx] = 0.f; denom[idx] = 0.f; amax[idx] = 0u; }
}

// K1: v_edge = W_edge @ att_edge  (tiny: one block of 128 threads)
__global__ void k_vedge(const float* __restrict__ W_edge,
                        const float* __restrict__ att_edge, float* __restrict__ v_edge) {
  int k = threadIdx.x;
  float s = 0.f;
  for (int j = 0; j < HID; ++j) s += W_edge[k * HID + j] * att_edge[j];
  v_edge[k] = s;
}

// K2: per real edge: a_edge = dot(edge_attr, v_edge); segment-sum a_edge and deg over dst.
// One wave32 per edge; lane owns a 16B chunk (wave reads 512B contiguous).
__global__ void k_edge(const float* __restrict__ edge_attr, const int* __restrict__ dst,
                       const float* __restrict__ v_edge, float* __restrict__ a_edge,
                       float* __restrict__ s_edge, float* __restrict__ deg, int E) {
  int wid  = (blockIdx.x * blockDim.x + threadIdx.x) >> 5;
  int lane = threadIdx.x & 31;
  if (wid >= E) return;
  float4 ea = *(const float4*)(edge_attr + (long long)wid * HID + lane * 4);
  float4 ve = *(const float4*)(v_edge + lane * 4);
  float p = ea.x * ve.x + ea.y * ve.y + ea.z * ve.z + ea.w * ve.w;
  for (int off = 16; off > 0; off >>= 1) p += __shfl_xor(p, off, 32);
  if (lane == 0) {
    a_edge[wid] = p;
    int d = dst[wid];
    atomicAdd(&s_edge[d], p);
    atomicAdd(&deg[d], 1.0f);
  }
}

// K3: h = x @ W via v_wmma_f32_16x16x4_f32.
// Block = 256 threads = 8 waves; block computes a 16x128 strip (wave w -> cols 16w..16w+15).
// A 16x4 f32 layout: lane M = lane&15, K base = (lane>>4)*2, VGPR0=K{0|2}, VGPR1=K{1|3}.
// B 4x16 mirrored; C/D 16x16 in 8 VGPRs (M = g or g+8, N = lane&15).
__global__ void k_gemm_h(const float* __restrict__ x, const float* __restrict__ W,
                         float* __restrict__ h, int N) {
  __shared__ float lx[16 * HID];
  int r0  = blockIdx.x * 16;
  int tid = threadIdx.x;
  for (int i = tid; i < 16 * HID; i += 256) {
    int r = r0 + (i >> 7);
    lx[i] = (r < N) ? x[(long long)r * HID + (i & (HID - 1))] : 0.f;
  }
  __syncthreads();
  int wv   = tid >> 5;          // col tile 0..7
  int lane = tid & 31;
  int m    = lane & 15;
  int kb   = (lane >> 4) * 2;   // 0 or 2
  int n    = wv * 16 + m;
  v8f acc = {};
  for (int k = 0; k < HID; k += 4) {
    v2f a, b;
    a.x = lx[m * HID + k + kb];
    a.y = lx[m * HID + k + kb + 1];
    b.x = W[(k + kb) * HID + n];
    b.y = W[(k + kb + 1) * HID + n];
    acc = __builtin_amdgcn_wmma_f32_16x16x4_f32(false, a, false, b, (short)0, acc,
                                                false, false);
  }
#pragma unroll
  for (int g = 0; g < 8; ++g) {
    int r = r0 + ((lane < 16) ? g : (g + 8));
    if (r < N) h[(long long)r * HID + n] = acc[g];
  }
}

// K4: a_src[i] = h[i].att_src ; a_dst[i] = h[i].att_dst  (one wave per node)
__global__ void k_attn_node(const float* __restrict__ h, const float* __restrict__ att_src,
                            const float* __restrict__ att_dst,
                            float* __restrict__ a_src, float* __restrict__ a_dst, int N) {
  int wid  = (blockIdx.x * blockDim.x + threadIdx.x) >> 5;
  int lane = threadIdx.x & 31;
  if (wid >= N) return;
  float4 hv = *(const float4*)(h + (long long)wid * HID + lane * 4);
  float4 as = *(const float4*)(att_src + lane * 4);
  float4 ad = *(const float4*)(att_dst + lane * 4);
  float ps = hv.x * as.x + hv.y * as.y + hv.z * as.z + hv.w * as.w;
  float pd = hv.x * ad.x + hv.y * ad.y + hv.z * ad.z + hv.w * ad.w;
  for (int off = 16; off > 0; off >>= 1) {
    ps += __shfl_xor(ps, off, 32);
    pd += __shfl_xor(pd, off, 32);
  }
  if (lane == 0) { a_src[wid] = ps; a_dst[wid] = pd; }
}

// K5: alpha = leakyrelu(a_src[s] + a_dst[d] + a_edge); segment-max via encoded atomicMax.
// Full edge list = E real edges + N self loops (self-loop a_edge = s_edge/max(deg,1)).
__global__ void k_alpha_max(const int* __restrict__ src, const int* __restrict__ dst,
                            const float* __restrict__ a_edge, const float* __restrict__ s_edge,
                            const float* __restrict__ deg, const float* __restrict__ a_src,
                            const float* __restrict__ a_dst, float* __restrict__ alpha,
                            unsigned* __restrict__ amax, int E, int N) {
  int idx = blockIdx.x * blockDim.x + threadIdx.x;
  if (idx >= E + N) return;
  int s, d; float ae;
  if (idx < E) { s = src[idx]; d = dst[idx]; ae = a_edge[idx]; }
  else { s = d = idx - E; ae = s_edge[d] / fmaxf(deg[d], 1.0f); }
  float al = a_src[s] + a_dst[d] + ae;
  al = (al >= 0.f) ? al : NEG_SLOPE * al;
  alpha[idx] = al;
  atomicMax(&amax[d], fkey(al));
}

// K6: ex = exp(alpha - amax[d]); denom segment-sum
__global__ void k_exp_sum(const int* __restrict__ dst, float* __restrict__ alpha,
                          const unsigned* __restrict__ amax, float* __restrict__ denom,
                          int E, int N) {
  int idx = blockIdx.x * blockDim.x + threadIdx.x;
  if (idx >= E + N) return;
  int d = (idx < E) ? dst[idx] : (idx - E);
  float ex = __expf(alpha[idx] - finv(amax[d]));
  alpha[idx] = ex;
  atomicAdd(&denom[d], ex);
}

// K7: out[d] += h[s] * (ex / (denom[d] + eps))  (one wave per full edge; f32 atomics to L2)
__global__ void k_scatter(const int* __restrict__ src, const int* __restrict__ dst,
                          const float* __restrict__ alpha, const float* __restrict__ denom,
                          const float* __restrict__ h, float* __restrict__ out, int E, int N) {
  int wid  = (blockIdx.x * blockDim.x + threadIdx.x) >> 5;
  int lane = threadIdx.x & 31;
  if (wid >= E + N) return;
  int s, d;
  if (wid < E) { s = src[wid]; d = dst[wid]; } else { s = d = wid - E; }
  float w = alpha[wid] / (denom[d] + GAT_EPS);
  float4 hv = *(const float4*)(h + (long long)s * HID + lane * 4);
  float* o = out + (long long)d * HID + lane * 4;
  atomicAdd(o + 0, hv.x * w);
  atomicAdd(o + 1, hv.y * w);
  atomicAdd(o + 2, hv.z * w);
  atomicAdd(o + 3, hv.w * w);
}

extern "C" void kernel_launch(void* const* d_in, const int* in_sizes, int n_in,
                              void* d_out, int out_size, void* d_ws, size_t ws_size,
                              hipStream_t stream) {
  (void)n_in; (void)out_size; (void)ws_size;
  const float* edge_attr = (const float*)d_in[0];
  const float* emb       = (const float*)d_in[1];
  const float* W         = (const float*)d_in[2];
  const float* W_edge    = (const float*)d_in[3];
  const float* att_src   = (const float*)d_in[4];
  const float* att_dst   = (const float*)d_in[5];
  const float* att_edge  = (const float*)d_in[6];
  const float* bias      = (const float*)d_in[7];
  const int*   eidx      = (const int*)d_in[8];

  const int E = in_sizes[0] / HID;   // edges from edge_attr rows
  const int N = in_sizes[1] / HID;   // nodes from emb_table rows (== entity_count)
  const int* src = eidx;
  const int* dst = eidx + E;

  // workspace carve-up (floats): h | a_edge | alpha | a_src | a_dst | s_edge | deg | denom | amax | v_edge
  float* p = (float*)d_ws;
  float* h       = p; p += (size_t)N * HID;
  float* a_edge  = p; p += E;
  float* alpha   = p; p += (size_t)E + N;
  float* a_src   = p; p += N;
  float* a_dst   = p; p += N;
  float* s_edge  = p; p += N;
  float* deg     = p; p += N;
  float* denom   = p; p += N;
  unsigned* amax = (unsigned*)p; p += N;
  float* v_edge  = p; p += HID;
  float* out = (float*)d_out;

  const long long tot = (long long)N * HID;
  const int F = E + N;

  k_init     <<<(int)((tot + 255) / 256), 256, 0, stream>>>(out, bias, s_edge, deg, denom, amax, N);
  k_vedge    <<<1, HID, 0, stream>>>(W_edge, att_edge, v_edge);
  k_edge     <<<(E + 7) / 8, 256, 0, stream>>>(edge_attr, dst, v_edge, a_edge, s_edge, deg, E);
  k_gemm_h   <<<(N + 15) / 16, 256, 0, stream>>>(emb, W, h, N);
  k_attn_node<<<(N + 7) / 8, 256, 0, stream>>>(h, att_src, att_dst, a_src, a_dst, N);
  k_alpha_max<<<(F + 255) / 256, 256, 0, stream>>>(src, dst, a_edge, s_edge, deg, a_src, a_dst,
                                                   alpha, amax, E, N);
  k_exp_sum  <<<(F + 255) / 256, 256, 0, stream>>>(dst, alpha, amax, denom, E, N);
  k_scatter  <<<(F + 7) / 8, 256, 0, stream>>>(src, dst, alpha, denom, h, out, E, N);
}